// RecyclingEmbedder_10737418240030
// MI455X (gfx1250) — compile-verified
//
#include <hip/hip_runtime.h>

typedef __attribute__((ext_vector_type(2))) float v2f;
typedef __attribute__((ext_vector_type(4))) float v4f;
typedef __attribute__((ext_vector_type(8))) float v8f;

#define INF_VAL 100000000.0f
#define LN_EPS 1e-5f

__device__ __forceinline__ float wave_reduce_add(float v) {
#pragma unroll
  for (int m = 16; m >= 1; m >>= 1) v += __shfl_xor(v, m, 32);
  return v;
}

// ---------------------------------------------------------------------------
// Kernel 1: pairwise distance bins via V_WMMA_F32_16X16X4_F32.
// G = X @ X^T computed 16x16 tile per wave32 block; K=4 (x,y,z,0 padded).
// d2 = |xi|^2 + |xj|^2 - 2*G[i][j]; bin = one-hot index into sq-bin ladder,
// -1 if no bin is hot. Written to workspace as int per (i,j).
// ---------------------------------------------------------------------------
__global__ __launch_bounds__(32)
void re_dist_bins_kernel(const float* __restrict__ x, int* __restrict__ bins,
                         int N) {
  const int lane = (int)threadIdx.x;
  const int i0 = (int)blockIdx.y * 16;
  const int j0 = (int)blockIdx.x * 16;
  const int half = lane >> 4;   // 0: K=0,1  1: K=2,3
  const int l = lane & 15;

  const int ia = (i0 + l) * 3;
  const int jb = (j0 + l) * 3;

  // Branchless A/B fill (EXEC must remain all-ones around WMMA).
  v2f a, b;
  a[0] = x[ia + (half ? 2 : 0)];
  a[1] = half ? 0.0f : x[ia + 1];
  b[0] = x[jb + (half ? 2 : 0)];
  b[1] = half ? 0.0f : x[jb + 1];

  v8f acc = {};
  // D = A(16x4) * B(4x16) + C : emits v_wmma_f32_16x16x4_f32
  acc = __builtin_amdgcn_wmma_f32_16x16x4_f32(
      /*neg_a=*/false, a, /*neg_b=*/false, b,
      /*c_mod=*/(short)0, acc, /*reuse_a=*/false, /*reuse_b=*/false);

  const int jcol = j0 + l;
  const float nj = x[jcol * 3 + 0] * x[jcol * 3 + 0] +
                   x[jcol * 3 + 1] * x[jcol * 3 + 1] +
                   x[jcol * 3 + 2] * x[jcol * 3 + 2];

#pragma unroll
  for (int r = 0; r < 8; ++r) {
    // C/D layout: VGPR r, lanes 0-15 -> M=r, lanes 16-31 -> M=r+8; N = lane&15
    const int irow = i0 + r + half * 8;
    const float ni = x[irow * 3 + 0] * x[irow * 3 + 0] +
                     x[irow * 3 + 1] * x[irow * 3 + 1] +
                     x[irow * 3 + 2] * x[irow * 3 + 2];
    const float d2 = ni + nj - 2.0f * acc[r];

    int bin = -1;
#pragma unroll
    for (int k = 0; k < 15; ++k) {
      const float bv = 3.25f + 1.25f * (float)k;
      const float lo = bv * bv;
      const float nv = 3.25f + 1.25f * (float)(k + 1);
      const float hi = (k < 14) ? (nv * nv) : INF_VAL;
      if (d2 > lo && d2 < hi) bin = k;
    }
    bins[(long long)irow * N + jcol] = bin;
  }
}

// ---------------------------------------------------------------------------
// Kernel 2: layernorm of m [N,256]. One wave32 per row, 8 floats per lane.
// ---------------------------------------------------------------------------
__global__ __launch_bounds__(256)
void re_m_ln_kernel(const float* __restrict__ m, const float* __restrict__ g,
                    const float* __restrict__ be, float* __restrict__ out,
                    int N) {
  const int tid = (int)threadIdx.x;
  const int wave = tid >> 5, lane = tid & 31;
  const int row = (int)blockIdx.x * 8 + wave;
  if (row >= N) return;

  const long long base = (long long)row * 256;
  const int c0 = lane * 4;
  const v4f a = *(const v4f*)(m + base + c0);
  const v4f b = *(const v4f*)(m + base + 128 + c0);

  float s = a[0] + a[1] + a[2] + a[3] + b[0] + b[1] + b[2] + b[3];
  float ss = a[0] * a[0] + a[1] * a[1] + a[2] * a[2] + a[3] * a[3] +
             b[0] * b[0] + b[1] * b[1] + b[2] * b[2] + b[3] * b[3];
  s = wave_reduce_add(s);
  ss = wave_reduce_add(ss);

  const float mu = s * (1.0f / 256.0f);
  const float var = ss * (1.0f / 256.0f) - mu * mu;
  const float inv = rsqrtf(var + LN_EPS);

  const v4f g0 = *(const v4f*)(g + c0);
  const v4f g1 = *(const v4f*)(g + 128 + c0);
  const v4f e0 = *(const v4f*)(be + c0);
  const v4f e1 = *(const v4f*)(be + 128 + c0);

  v4f o0, o1;
#pragma unroll
  for (int t = 0; t < 4; ++t) {
    o0[t] = (a[t] - mu) * inv * g0[t] + e0[t];
    o1[t] = (b[t] - mu) * inv * g1[t] + e1[t];
  }
  *(v4f*)(out + base + c0) = o0;
  *(v4f*)(out + base + 128 + c0) = o1;
}

// ---------------------------------------------------------------------------
// Kernel 3 (bandwidth-dominant): stream z [N*N,128], layernorm each row in a
// single wave32 (float4 per lane), add w_lin[:,bin] (from LDS) + b_lin.
// Nontemporal 128-bit loads/stores: 604 MB traffic >> 192 MB L2.
// ---------------------------------------------------------------------------
__global__ __launch_bounds__(256)
void re_z_ln_kernel(const float* __restrict__ z, const float* __restrict__ gz,
                    const float* __restrict__ bz,
                    const float* __restrict__ w_lin,
                    const float* __restrict__ b_lin,
                    const int* __restrict__ bins, float* __restrict__ out,
                    long long nrows) {
  // s_w[bin*128 + c] = w_lin[c*15 + bin]; row 15 is all-zero (bin = -1).
  __shared__ float s_w[16 * 128];
  const int tid = (int)threadIdx.x;
#pragma unroll
  for (int i = tid; i < 2048; i += 256) {
    const int b = i >> 7, c = i & 127;
    s_w[i] = (b < 15) ? w_lin[c * 15 + b] : 0.0f;
  }
  __syncthreads();

  const int wave = tid >> 5, lane = tid & 31;
  const int c0 = lane * 4;
  const v4f g4 = *(const v4f*)(gz + c0);
  const v4f e4 = *(const v4f*)(bz + c0);
  const v4f bl4 = *(const v4f*)(b_lin + c0);

  const long long wglobal = (long long)blockIdx.x * 8 + wave;

#pragma unroll
  for (int k = 0; k < 8; ++k) {
    const long long row = wglobal * 8 + (long long)k;
    if (row >= nrows) break;

    const v4f v = __builtin_nontemporal_load((const v4f*)(z + row * 128 + c0));

    float s = v[0] + v[1] + v[2] + v[3];
    float ss = v[0] * v[0] + v[1] * v[1] + v[2] * v[2] + v[3] * v[3];
    s = wave_reduce_add(s);
    ss = wave_reduce_add(ss);

    const float mu = s * (1.0f / 128.0f);
    const float var = ss * (1.0f / 128.0f) - mu * mu;
    const float inv = rsqrtf(var + LN_EPS);

    const int bn = bins[row];
    const int woff = ((bn < 0) ? 15 : bn) * 128 + c0;

    v4f o;
#pragma unroll
    for (int t = 0; t < 4; ++t)
      o[t] = (v[t] - mu) * inv * g4[t] + e4[t] + s_w[woff + t] + bl4[t];

    __builtin_nontemporal_store(o, (v4f*)(out + row * 128 + c0));
  }
}

// ---------------------------------------------------------------------------
extern "C" void kernel_launch(void* const* d_in, const int* in_sizes, int n_in,
                              void* d_out, int out_size, void* d_ws,
                              size_t ws_size, hipStream_t stream) {
  const float* m = (const float*)d_in[0];
  const float* z = (const float*)d_in[1];
  const float* x = (const float*)d_in[2];
  const float* w_lin = (const float*)d_in[3];
  const float* b_lin = (const float*)d_in[4];
  const float* g_m = (const float*)d_in[5];
  const float* be_m = (const float*)d_in[6];
  const float* g_z = (const float*)d_in[7];
  const float* be_z = (const float*)d_in[8];
  (void)n_in; (void)out_size; (void)ws_size;

  const int N = in_sizes[0] / 256;  // 768

  float* m_out = (float*)d_out;
  float* z_out = (float*)d_out + (long long)N * 256;
  int* bins = (int*)d_ws;  // N*N ints (2.36 MB for N=768)

  // 1) WMMA Gram-matrix -> distance bins
  dim3 gbin((unsigned)(N / 16), (unsigned)(N / 16));
  re_dist_bins_kernel<<<gbin, 32, 0, stream>>>(x, bins, N);

  // 2) m layernorm
  re_m_ln_kernel<<<(N + 7) / 8, 256, 0, stream>>>(m, g_m, be_m, m_out, N);

  // 3) z layernorm + one-hot linear add (streaming, bandwidth-bound)
  const long long nrows = (long long)N * N;
  const long long nwaves = (nrows + 7) / 8;
  const int nblocks = (int)((nwaves + 7) / 8);
  re_z_ln_kernel<<<nblocks, 256, 0, stream>>>(z, g_z, be_z, w_lin, b_lin, bins,
                                              z_out, nrows);
}